// Baller2Vec_36163624633038
// MI455X (gfx1250) — compile-verified
//
#include <hip/hip_runtime.h>
#include <hip/hip_bf16.h>

// ---------------------------------------------------------------------------
// Baller2Vec forward: input MLP -> 6x post-norm transformer layers -> head.
// fp32 end-to-end. GEMMs (QKV/O projections + FFN, ~92% of FLOPs) run on
// V_WMMA_F32_16X16X4_F32: 64x16 output strip per wave (B-fragment reuse),
// clamped unconditional loads, 2-stage software-pipelined K loop with
// sched_group_barrier pinning (load-batch || WMMA-quad overlap).
// Attention / LN / pool / head-GEMV / softmax are VALU (skinny shapes).
// ---------------------------------------------------------------------------

typedef __attribute__((ext_vector_type(2))) float v2f;
typedef __attribute__((ext_vector_type(8))) float v8f;

#define SEQ     184          // tokens
#define EMB     184          // d_model
#define NHEAD   8
#define HD      23           // head_dim
#define DFF     2048
#define NLAYERS 6
#define NPLAB   7140
#define NPLAYERS 22

#if __has_builtin(__builtin_amdgcn_sched_group_barrier)
#define SCHED_VMEM(n)  __builtin_amdgcn_sched_group_barrier(0x020, (n), 0)  // VMEM read
#define SCHED_WMMA(n)  __builtin_amdgcn_sched_group_barrier(0x008, (n), 0)  // MFMA/WMMA
#else
#define SCHED_VMEM(n)
#define SCHED_WMMA(n)
#endif

#define WMMA4(x0, x1, x2, x3, bb)                                              \
    do {                                                                       \
        acc0 = __builtin_amdgcn_wmma_f32_16x16x4_f32(false, x0, false, bb,     \
                                                     (short)0, acc0, false, false); \
        acc1 = __builtin_amdgcn_wmma_f32_16x16x4_f32(false, x1, false, bb,     \
                                                     (short)0, acc1, false, false); \
        acc2 = __builtin_amdgcn_wmma_f32_16x16x4_f32(false, x2, false, bb,     \
                                                     (short)0, acc2, false, false); \
        acc3 = __builtin_amdgcn_wmma_f32_16x16x4_f32(false, x3, false, bb,     \
                                                     (short)0, acc3, false, false); \
    } while (0)

// ---------------------------------------------------------------------------
// WMMA f32 GEMM: C[m,n] = sum_k A[m,k] * W[n,k]  (+bias[n]) (+R[m,n]) (relu)
// A: [184, K] row-major, W: [N, K] row-major, C: [184, N].
// One wave computes a 64x16 strip = 4 stacked 16x16 tiles sharing B fragments.
// K is a multiple of 8 (184 and 2048 both are). OOB rows/cols are CLAMPED to
// row 0 (their products land only in never-stored accumulator slots), so all
// loads are unconditional. K loop is double-buffered; sched_group_barriers
// enforce (5 loads)(4 wmma)(5 loads)(4 wmma) so each load batch flies while
// the previous batch's WMMAs execute.
// ---------------------------------------------------------------------------
__global__ __launch_bounds__(256)
void gemm_wmma_f32(const float* __restrict__ A,
                   const float* __restrict__ W,
                   const float* __restrict__ bias,   // may be null
                   const float* __restrict__ R,      // residual, may be null
                   float* __restrict__ C,
                   int N, int K, int relu)
{
    const int lane = threadIdx.x & 31;
    const int wave = threadIdx.x >> 5;
    const int ntiles = (N + 15) >> 4;
    const int tile = blockIdx.x * 8 + wave;    // 3 M-groups x ntiles
    if (tile >= 3 * ntiles) return;            // uniform per wave
    const int tg = tile / ntiles;              // rows [tg*64, tg*64+64)
    const int tn = tile % ntiles;

    const int hlf = lane >> 4;                 // 0: K+0/1, 1: K+2/3
    const int l15 = lane & 15;
    const int n = tn * 16 + l15;               // output col held by this lane
    const int ncl = (n < N) ? n : 0;           // clamped W row

    // 32-bit fragment offsets (uniform base + per-lane offset addressing)
    int aoff0, aoff1, aoff2, aoff3;
    {
        int m0 = tg * 64 + l15;
        int m1 = m0 + 16, m2 = m0 + 32, m3 = m0 + 48;
        if (m0 >= SEQ) m0 = 0;
        if (m1 >= SEQ) m1 = 0;
        if (m2 >= SEQ) m2 = 0;
        if (m3 >= SEQ) m3 = 0;
        aoff0 = m0 * K + hlf * 2;
        aoff1 = m1 * K + hlf * 2;
        aoff2 = m2 * K + hlf * 2;
        aoff3 = m3 * K + hlf * 2;
    }
    const int boff = ncl * K + hlf * 2;

    v8f acc0 = {}, acc1 = {}, acc2 = {}, acc3 = {};

    // stage X: preload k=0
    v2f bX = *(const v2f*)(W + boff  + 0);
    v2f x0 = *(const v2f*)(A + aoff0 + 0);
    v2f x1 = *(const v2f*)(A + aoff1 + 0);
    v2f x2 = *(const v2f*)(A + aoff2 + 0);
    v2f x3 = *(const v2f*)(A + aoff3 + 0);

    int k0 = 0;
    for (; k0 + 8 < K; k0 += 8) {
        // stage Y loads (k0+4) in flight while X WMMAs run
        v2f bY = *(const v2f*)(W + boff  + k0 + 4);
        v2f y0 = *(const v2f*)(A + aoff0 + k0 + 4);
        v2f y1 = *(const v2f*)(A + aoff1 + k0 + 4);
        v2f y2 = *(const v2f*)(A + aoff2 + k0 + 4);
        v2f y3 = *(const v2f*)(A + aoff3 + k0 + 4);
        WMMA4(x0, x1, x2, x3, bX);
        // stage X loads (k0+8) in flight while Y WMMAs run
        bX = *(const v2f*)(W + boff  + k0 + 8);
        x0 = *(const v2f*)(A + aoff0 + k0 + 8);
        x1 = *(const v2f*)(A + aoff1 + k0 + 8);
        x2 = *(const v2f*)(A + aoff2 + k0 + 8);
        x3 = *(const v2f*)(A + aoff3 + k0 + 8);
        WMMA4(y0, y1, y2, y3, bY);
        // pin the pipeline: loads batch ahead of the wmma quad that follows
        SCHED_VMEM(5);
        SCHED_WMMA(4);
        SCHED_VMEM(5);
        SCHED_WMMA(4);
    }
    // tail: exactly 8 K-values remain (K % 8 == 0)
    {
        v2f bY = *(const v2f*)(W + boff  + k0 + 4);
        v2f y0 = *(const v2f*)(A + aoff0 + k0 + 4);
        v2f y1 = *(const v2f*)(A + aoff1 + k0 + 4);
        v2f y2 = *(const v2f*)(A + aoff2 + k0 + 4);
        v2f y3 = *(const v2f*)(A + aoff3 + k0 + 4);
        WMMA4(x0, x1, x2, x3, bX);
        WMMA4(y0, y1, y2, y3, bY);
        SCHED_VMEM(5);
        SCHED_WMMA(8);
    }

    const bool nv = (n < N);
    const float bn = (bias != nullptr && nv) ? bias[n] : 0.f;
    v8f accs[4] = {acc0, acc1, acc2, acc3};
    #pragma unroll
    for (int t = 0; t < 4; ++t) {
        #pragma unroll
        for (int r = 0; r < 8; ++r) {
            const int mr = tg * 64 + t * 16 + r + hlf * 8;  // C/D row layout
            if (mr < SEQ && nv) {
                float v = accs[t][r] + bn;
                if (R) v += R[mr * N + n];
                if (relu) v = fmaxf(v, 0.f);
                C[mr * N + n] = v;
            }
        }
    }
}

// ---------------------------------------------------------------------------
// Input MLP: x[184,3] -> 64 -> 128 -> 184, ReLU after last, * sqrt(184).
// One block per row.
// ---------------------------------------------------------------------------
__global__ __launch_bounds__(64)
void mlp_kernel(const float* __restrict__ x,
                const float* __restrict__ w0, const float* __restrict__ b0,
                const float* __restrict__ w1, const float* __restrict__ b1,
                const float* __restrict__ w2, const float* __restrict__ b2,
                float* __restrict__ h)
{
    __shared__ float t0[64];
    __shared__ float t1[128];
    const int row = blockIdx.x;       // 0..183
    const int t = threadIdx.x;        // 0..63
    const float x0 = x[row * 3 + 0];
    const float x1 = x[row * 3 + 1];
    const float x2 = x[row * 3 + 2];
    t0[t] = w0[t * 3 + 0] * x0 + w0[t * 3 + 1] * x1 + w0[t * 3 + 2] * x2 + b0[t];
    __syncthreads();
    for (int o = t; o < 128; o += 64) {
        float a = b1[o];
        for (int i = 0; i < 64; ++i) a += w1[o * 64 + i] * t0[i];
        t1[o] = a;
    }
    __syncthreads();
    for (int o = t; o < EMB; o += 64) {
        float a = b2[o];
        for (int i = 0; i < 128; ++i) a += w2[o * 128 + i] * t1[i];
        h[row * EMB + o] = fmaxf(a, 0.f) * 13.564659966250536f;  // sqrt(184)
    }
}

// ---------------------------------------------------------------------------
// Attention: wave per (head, query). Block-causal mask on the step index.
// ---------------------------------------------------------------------------
__device__ __forceinline__ int token_step(int i) {
    return (i < NPLAYERS * 8) ? (i / NPLAYERS) : (i - NPLAYERS * 8);
}

__global__ __launch_bounds__(256)
void attn_kernel(const float* __restrict__ Q, const float* __restrict__ K,
                 const float* __restrict__ V, float* __restrict__ ctx)
{
    const int lane = threadIdx.x & 31;
    const int wave = threadIdx.x >> 5;
    const int idx = blockIdx.x * 8 + wave;     // head * SEQ + q
    if (idx >= NHEAD * SEQ) return;
    const int hh = idx / SEQ;
    const int q  = idx % SEQ;
    const int qs = token_step(q);

    float qv[HD];
    const float* qrow = Q + (size_t)q * EMB + hh * HD;
    #pragma unroll
    for (int d = 0; d < HD; ++d) qv[d] = qrow[d];

    const float scale = 0.20851441405707477f;  // 1/sqrt(23)
    float sc[6];
    bool  ok[6];
    float mx = -3.4e38f;
    for (int c = 0; c < 6; ++c) {
        const int j = lane + 32 * c;
        float s = -3.4e38f;
        bool valid = (j < SEQ) && (token_step(j) <= qs);
        if (valid) {
            const float* krow = K + (size_t)j * EMB + hh * HD;
            float acc = 0.f;
            #pragma unroll
            for (int d = 0; d < HD; ++d) acc += qv[d] * krow[d];
            s = acc * scale;
        }
        sc[c] = s; ok[c] = valid;
        mx = fmaxf(mx, s);
    }
    for (int off = 16; off; off >>= 1) mx = fmaxf(mx, __shfl_xor(mx, off, 32));

    float sum = 0.f;
    for (int c = 0; c < 6; ++c) {
        float e = ok[c] ? __expf(sc[c] - mx) : 0.f;
        sc[c] = e; sum += e;
    }
    for (int off = 16; off; off >>= 1) sum += __shfl_xor(sum, off, 32);
    const float inv = 1.f / sum;

    float cacc[HD];
    #pragma unroll
    for (int d = 0; d < HD; ++d) cacc[d] = 0.f;
    for (int c = 0; c < 6; ++c) {
        const int j = lane + 32 * c;
        if (ok[c]) {
            const float p = sc[c] * inv;
            const float* vrow = V + (size_t)j * EMB + hh * HD;
            #pragma unroll
            for (int d = 0; d < HD; ++d) cacc[d] += p * vrow[d];
        }
    }
    #pragma unroll
    for (int d = 0; d < HD; ++d) {
        float t = cacc[d];
        for (int off = 16; off; off >>= 1) t += __shfl_xor(t, off, 32);
        if (lane == 0) ctx[(size_t)q * EMB + hh * HD + d] = t;
    }
}

// ---------------------------------------------------------------------------
// LayerNorm: wave per row of [184,184].
// ---------------------------------------------------------------------------
__global__ __launch_bounds__(256)
void ln_kernel(const float* __restrict__ X, const float* __restrict__ g,
               const float* __restrict__ b, float* __restrict__ Y)
{
    const int lane = threadIdx.x & 31;
    const int wave = threadIdx.x >> 5;
    const int row = blockIdx.x * 8 + wave;
    if (row >= SEQ) return;
    const float* x = X + (size_t)row * EMB;

    float vals[6];
    float s = 0.f;
    for (int c = 0; c < 6; ++c) {
        const int j = lane + 32 * c;
        float v = (j < EMB) ? x[j] : 0.f;
        vals[c] = v; s += v;
    }
    for (int off = 16; off; off >>= 1) s += __shfl_xor(s, off, 32);
    const float mu = s * (1.f / EMB);

    float var = 0.f;
    for (int c = 0; c < 6; ++c) {
        const int j = lane + 32 * c;
        float d = (j < EMB) ? (vals[c] - mu) : 0.f;
        var += d * d;
    }
    for (int off = 16; off; off >>= 1) var += __shfl_xor(var, off, 32);
    const float inv = rsqrtf(var * (1.f / EMB) + 1e-5f);

    for (int c = 0; c < 6; ++c) {
        const int j = lane + 32 * c;
        if (j < EMB) Y[(size_t)row * EMB + j] = (vals[c] - mu) * inv * g[j] + b[j];
    }
}

// ---------------------------------------------------------------------------
// pooled[row] = mean_j h[row, j]
// ---------------------------------------------------------------------------
__global__ __launch_bounds__(256)
void pool_kernel(const float* __restrict__ h, float* __restrict__ pooled)
{
    const int lane = threadIdx.x & 31;
    const int wave = threadIdx.x >> 5;
    const int row = blockIdx.x * 8 + wave;
    if (row >= SEQ) return;
    float s = 0.f;
    for (int c = 0; c < 6; ++c) {
        const int j = lane + 32 * c;
        if (j < EMB) s += h[(size_t)row * EMB + j];
    }
    for (int off = 16; off; off >>= 1) s += __shfl_xor(s, off, 32);
    if (lane == 0) pooled[row] = s * (1.f / EMB);
}

// ---------------------------------------------------------------------------
// Head GEMV: out[row] = sum_j W[row, j] * pooled[j] + bias[row]. Wave per row.
// This is the bandwidth-dominant piece (pc_w = 115.6 MB fp32, read once).
// ---------------------------------------------------------------------------
__global__ __launch_bounds__(256)
void gemv_kernel(const float* __restrict__ W, const float* __restrict__ bias,
                 const float* __restrict__ pooled, float* __restrict__ out, int N)
{
    const int lane = threadIdx.x & 31;
    const int wave = threadIdx.x >> 5;
    const int row = blockIdx.x * 8 + wave;
    if (row >= N) return;
    const float* w = W + (size_t)row * EMB;
    float s = 0.f;
    for (int c = 0; c < 6; ++c) {
        const int j = lane + 32 * c;
        if (j < EMB) s += w[j] * pooled[j];
    }
    for (int off = 16; off; off >>= 1) s += __shfl_xor(s, off, 32);
    if (lane == 0) out[row] = s + bias[row];
}

// ---------------------------------------------------------------------------
// Softmax over 23 rows of length 7140 (row 0 = ball, rows 1..22 = players).
// Logits layout in ws matches d_out layout, so src/dst share indexing.
// ---------------------------------------------------------------------------
__global__ __launch_bounds__(256)
void softmax_kernel(const float* __restrict__ logits, float* __restrict__ out)
{
    const int row = blockIdx.x;
    const float* src = logits + (size_t)row * NPLAB;
    float* dst = out + (size_t)row * NPLAB;
    __shared__ float red[8];
    __shared__ float bmax, bsum;
    const int t = threadIdx.x, lane = t & 31, wave = t >> 5;

    float mx = -3.4e38f;
    for (int j = t; j < NPLAB; j += 256) mx = fmaxf(mx, src[j]);
    for (int off = 16; off; off >>= 1) mx = fmaxf(mx, __shfl_xor(mx, off, 32));
    if (lane == 0) red[wave] = mx;
    __syncthreads();
    if (t == 0) {
        float v = red[0];
        for (int w = 1; w < 8; ++w) v = fmaxf(v, red[w]);
        bmax = v;
    }
    __syncthreads();
    mx = bmax;

    float sum = 0.f;
    for (int j = t; j < NPLAB; j += 256) sum += __expf(src[j] - mx);
    for (int off = 16; off; off >>= 1) sum += __shfl_xor(sum, off, 32);
    if (lane == 0) red[wave] = sum;
    __syncthreads();
    if (t == 0) {
        float v = 0.f;
        for (int w = 0; w < 8; ++w) v += red[w];
        bsum = v;
    }
    __syncthreads();
    const float inv = 1.f / bsum;
    for (int j = t; j < NPLAB; j += 256) dst[j] = __expf(src[j] - mx) * inv;
}

// ---------------------------------------------------------------------------
// Host-side launcher.
// ---------------------------------------------------------------------------
extern "C" void kernel_launch(void* const* d_in, const int* in_sizes, int n_in,
                              void* d_out, int out_size, void* d_ws, size_t ws_size,
                              hipStream_t stream)
{
    (void)in_sizes; (void)n_in; (void)out_size; (void)ws_size;
    const float* x      = (const float*)d_in[0];
    const float* mlp_w0 = (const float*)d_in[1];
    const float* mlp_b0 = (const float*)d_in[2];
    const float* mlp_w1 = (const float*)d_in[3];
    const float* mlp_b1 = (const float*)d_in[4];
    const float* mlp_w2 = (const float*)d_in[5];
    const float* mlp_b2 = (const float*)d_in[6];
    const float* wq     = (const float*)d_in[7];
    const float* bq     = (const float*)d_in[8];
    const float* wk     = (const float*)d_in[9];
    const float* bk     = (const float*)d_in[10];
    const float* wv     = (const float*)d_in[11];
    const float* bv     = (const float*)d_in[12];
    const float* wo     = (const float*)d_in[13];
    const float* bo     = (const float*)d_in[14];
    const float* ln1_g  = (const float*)d_in[15];
    const float* ln1_b  = (const float*)d_in[16];
    const float* w1     = (const float*)d_in[17];
    const float* b1     = (const float*)d_in[18];
    const float* w2     = (const float*)d_in[19];
    const float* b2     = (const float*)d_in[20];
    const float* ln2_g  = (const float*)d_in[21];
    const float* ln2_b  = (const float*)d_in[22];
    const float* pc_w   = (const float*)d_in[23];
    const float* pc_b   = (const float*)d_in[24];
    const float* bc_w   = (const float*)d_in[25];
    const float* bc_b   = (const float*)d_in[26];

    // workspace layout (floats)
    float* ws     = (float*)d_ws;
    float* h      = ws;                      // 184*184
    float* qb     = h    + SEQ * EMB;
    float* kb     = qb   + SEQ * EMB;
    float* vb     = kb   + SEQ * EMB;
    float* ctx    = vb   + SEQ * EMB;
    float* tmp    = ctx  + SEQ * EMB;        // pre-LN scratch
    float* a1     = tmp  + SEQ * EMB;        // 184*2048
    float* pooled = a1   + SEQ * DFF;        // 184
    float* logits = pooled + SEQ;            // 7140 + 157080 (ball first)

    dim3 blk(256);
    const int gemm184  = 5;    // 3*12 tiles / 8 waves
    const int gemm2048 = 48;   // 3*128 tiles / 8 waves
    const int rows8 = 23;      // 184 rows / 8 waves

    mlp_kernel<<<SEQ, 64, 0, stream>>>(x, mlp_w0, mlp_b0, mlp_w1, mlp_b1,
                                       mlp_w2, mlp_b2, h);

    for (int l = 0; l < NLAYERS; ++l) {
        const float* Wq = wq + (size_t)l * EMB * EMB;
        const float* Wk = wk + (size_t)l * EMB * EMB;
        const float* Wv = wv + (size_t)l * EMB * EMB;
        const float* Wo = wo + (size_t)l * EMB * EMB;
        const float* W1 = w1 + (size_t)l * DFF * EMB;
        const float* W2 = w2 + (size_t)l * EMB * DFF;

        gemm_wmma_f32<<<gemm184, blk, 0, stream>>>(h, Wq, bq + l * EMB, nullptr, qb, EMB, EMB, 0);
        gemm_wmma_f32<<<gemm184, blk, 0, stream>>>(h, Wk, bk + l * EMB, nullptr, kb, EMB, EMB, 0);
        gemm_wmma_f32<<<gemm184, blk, 0, stream>>>(h, Wv, bv + l * EMB, nullptr, vb, EMB, EMB, 0);
        attn_kernel<<<SEQ, blk, 0, stream>>>(qb, kb, vb, ctx);   // 8*184 waves
        gemm_wmma_f32<<<gemm184, blk, 0, stream>>>(ctx, Wo, bo + l * EMB, h, tmp, EMB, EMB, 0);
        ln_kernel<<<rows8, blk, 0, stream>>>(tmp, ln1_g + l * EMB, ln1_b + l * EMB, h);
        gemm_wmma_f32<<<gemm2048, blk, 0, stream>>>(h, W1, b1 + l * DFF, nullptr, a1, DFF, EMB, 1);
        gemm_wmma_f32<<<gemm184, blk, 0, stream>>>(a1, W2, b2 + l * EMB, h, tmp, EMB, DFF, 0);
        ln_kernel<<<rows8, blk, 0, stream>>>(tmp, ln2_g + l * EMB, ln2_b + l * EMB, h);
    }

    pool_kernel<<<rows8, blk, 0, stream>>>(h, pooled);
    // ball logits first (matches output layout), then player logits
    gemv_kernel<<<(NPLAB + 7) / 8, blk, 0, stream>>>(bc_w, bc_b, pooled, logits, NPLAB);
    gemv_kernel<<<(NPLAB * NPLAYERS + 7) / 8, blk, 0, stream>>>(
        pc_w, pc_b, pooled, logits + NPLAB, NPLAB * NPLAYERS);
    softmax_kernel<<<1 + NPLAYERS, blk, 0, stream>>>(logits, (float*)d_out);
}